// GNN_54752243089879
// MI455X (gfx1250) — compile-verified
//
#include <hip/hip_runtime.h>

typedef float v2f __attribute__((ext_vector_type(2)));
typedef float v8f __attribute__((ext_vector_type(8)));

#define DIM 256
#define WPITCH 260   // LDS row pitch (floats): 260 % 64 == 4 -> conflict-free b64 reads across 16 lanes

// ---------------- degree / dinv ----------------

__global__ __launch_bounds__(256) void deg_init_kernel(float* __restrict__ deg, int n) {
    int i = blockIdx.x * 256 + threadIdx.x;
    if (i < n) deg[i] = 1.0f;  // self-loop contributes 1 to every node's degree
}

__global__ __launch_bounds__(256) void deg_accum_kernel(const int* __restrict__ row,
                                                        float* deg, int e) {
    int i = blockIdx.x * 256 + threadIdx.x;
    if (i < e) atomicAdd(&deg[row[i]], 1.0f);
}

__global__ __launch_bounds__(256) void dinv_kernel(float* __restrict__ deg, int n) {
    int i = blockIdx.x * 256 + threadIdx.x;
    if (i < n) deg[i] = rsqrtf(deg[i]);  // deg >= 1 always
}

// ---------------- WMMA f32 GEMM + bias ----------------
// OUT[nrows][ncols] = A[nrows][256] @ W[256][ncols] + bias
// Block = 256 threads (8 waves), covers 128 rows x (16*CT) cols.
// Each wave computes a 16 x (16*CT) strip: CT accumulators, A fragment reused CT times.
// grid = (ceil(nrows/128), ncols/(16*CT)).

template <int CT>
__global__ __launch_bounds__(256)
void gemm_bias_wmma(const float* __restrict__ A, const float* __restrict__ W,
                    const float* __restrict__ bias, float* __restrict__ OUT,
                    int nrows, int ncols)
{
    __shared__ float wlds[16 * CT * WPITCH];  // transposed panel: wlds[n*WPITCH + k] = W[k][colBase+n]
    const int tid = threadIdx.x;
    const int colBase = blockIdx.y * (16 * CT);

    for (int i = tid; i < 16 * CT * DIM; i += 256) {
        int k = i / (16 * CT);
        int n = i % (16 * CT);
        wlds[n * WPITCH + k] = W[(size_t)k * ncols + colBase + n];
    }
    __syncthreads();

    const int wave = tid >> 5;
    const int lane = tid & 31;
    const int rowBase = blockIdx.x * 128 + wave * 16;
    if (rowBase >= nrows) return;       // wave-uniform: EXEC stays all-1s for WMMA

    const int lrow = lane & 15;         // M (A) / N (B) index within 16-tile
    const int hi   = (lane >> 4) << 1;  // K sub-offset: lanes 0-15 -> K+{0,1}, lanes 16-31 -> K+{2,3}

    const float* __restrict__ arow = A + (size_t)(rowBase + lrow) * DIM + hi;  // 8B aligned
    const float* __restrict__ brow = wlds + lrow * WPITCH + hi;                // 8B aligned

    v8f acc[CT];
#pragma unroll
    for (int c = 0; c < CT; ++c) acc[c] = (v8f){};

#pragma unroll
    for (int k = 0; k < DIM; k += 4) {
        v2f a = *(const v2f*)(arow + k);            // one global_load_b64, reused CT times
#pragma unroll
        for (int c = 0; c < CT; ++c) {
            v2f b = *(const v2f*)(brow + c * 16 * WPITCH + k);  // ds_load_b64
            acc[c] = __builtin_amdgcn_wmma_f32_16x16x4_f32(false, a, false, b,
                                                           (short)0, acc[c], false, false);
        }
    }

    // C/D layout: vgpr i, lane L -> M = i + 8*(L>=16), N = L%16
#pragma unroll
    for (int c = 0; c < CT; ++c) {
        const int colc = colBase + c * 16 + lrow;
        const float bc = bias[colc];
        float* __restrict__ orow = OUT + (size_t)(rowBase + (hi << 2)) * ncols + colc;
#pragma unroll
        for (int i = 0; i < 8; ++i)
            orow[(size_t)i * ncols] = acc[c][i] + bc;
    }
}

// ---------------- APPNP propagation ----------------

// AGG[i] = dinv[i]^2 * NB[i]   (self-loop term). 64 lanes (float4) per row.
__global__ __launch_bounds__(256)
void agg_init_kernel(const float* __restrict__ NB, const float* __restrict__ dinv,
                     float* __restrict__ AGG, int n)
{
    int idx = blockIdx.x * 256 + threadIdx.x;
    int r = idx >> 6;
    if (r >= n) return;
    int l = (idx & 63) << 2;
    float w = dinv[r] * dinv[r];
    const float4 v = *(const float4*)(NB + (size_t)r * DIM + l);
    float4 o;
    o.x = w * v.x; o.y = w * v.y; o.z = w * v.z; o.w = w * v.w;
    *(float4*)(AGG + (size_t)r * DIM + l) = o;
}

// AGG[row[e]] += dinv[row]*dinv[col] * NB[col[e]].  64 lanes (float4) per edge.
__global__ __launch_bounds__(256)
void edge_scatter_kernel(const int* __restrict__ row, const int* __restrict__ col,
                         const float* __restrict__ dinv, const float* __restrict__ NB,
                         float* AGG, int e)
{
    int idx = blockIdx.x * 256 + threadIdx.x;
    int ei = idx >> 6;
    if (ei >= e) return;
    int l = (idx & 63) << 2;
    int r = row[ei], c = col[ei];
    float w = dinv[r] * dinv[c];
    const float4 v = *(const float4*)(NB + (size_t)c * DIM + l);
    float* dst = AGG + (size_t)r * DIM + l;
    atomicAdd(dst + 0, w * v.x);
    atomicAdd(dst + 1, w * v.y);
    atomicAdd(dst + 2, w * v.z);
    atomicAdd(dst + 3, w * v.w);
}

// H = relu( l2norm_row( T + 0.5*AGG + 0.5*NB ) )   -- one wave32 per row, 8 floats/lane.
__global__ __launch_bounds__(256)
void combine_norm_relu_kernel(const float* __restrict__ T, const float* __restrict__ NB,
                              const float* __restrict__ AGG, float* __restrict__ H, int n)
{
    int r = blockIdx.x * 8 + (threadIdx.x >> 5);
    if (r >= n) return;
    int lane = threadIdx.x & 31;
    size_t base = (size_t)r * DIM + lane * 8;

    float4 t0 = *(const float4*)(T + base);
    float4 t1 = *(const float4*)(T + base + 4);
    float4 b0 = *(const float4*)(NB + base);
    float4 b1 = *(const float4*)(NB + base + 4);
    float4 g0 = *(const float4*)(AGG + base);
    float4 g1 = *(const float4*)(AGG + base + 4);

    float v[8];
    v[0] = t0.x + 0.5f * (g0.x + b0.x);
    v[1] = t0.y + 0.5f * (g0.y + b0.y);
    v[2] = t0.z + 0.5f * (g0.z + b0.z);
    v[3] = t0.w + 0.5f * (g0.w + b0.w);
    v[4] = t1.x + 0.5f * (g1.x + b1.x);
    v[5] = t1.y + 0.5f * (g1.y + b1.y);
    v[6] = t1.z + 0.5f * (g1.z + b1.z);
    v[7] = t1.w + 0.5f * (g1.w + b1.w);

    float s = 0.0f;
#pragma unroll
    for (int i = 0; i < 8; ++i) s += v[i] * v[i];
#pragma unroll
    for (int off = 16; off > 0; off >>= 1) s += __shfl_xor(s, off, 32);

    float inv = 1.0f / fmaxf(sqrtf(s), 1e-12f);
    float4 o0, o1;
    o0.x = fmaxf(v[0] * inv, 0.0f); o0.y = fmaxf(v[1] * inv, 0.0f);
    o0.z = fmaxf(v[2] * inv, 0.0f); o0.w = fmaxf(v[3] * inv, 0.0f);
    o1.x = fmaxf(v[4] * inv, 0.0f); o1.y = fmaxf(v[5] * inv, 0.0f);
    o1.z = fmaxf(v[6] * inv, 0.0f); o1.w = fmaxf(v[7] * inv, 0.0f);
    *(float4*)(H + base) = o0;
    *(float4*)(H + base + 4) = o1;
}

// ---------------- driver ----------------

extern "C" void kernel_launch(void* const* d_in, const int* in_sizes, int n_in,
                              void* d_out, int out_size, void* d_ws, size_t ws_size,
                              hipStream_t stream)
{
    (void)n_in; (void)out_size; (void)ws_size;

    const float* x   = (const float*)d_in[0];
    const int*   ei  = (const int*)d_in[1];
    const float* W0  = (const float*)d_in[2];
    const float* b0  = (const float*)d_in[3];
    const float* Wn0 = (const float*)d_in[4];
    const float* bn0 = (const float*)d_in[5];
    const float* W1  = (const float*)d_in[6];
    const float* b1  = (const float*)d_in[7];
    const float* Wn1 = (const float*)d_in[8];
    const float* bn1 = (const float*)d_in[9];
    const float* W2  = (const float*)d_in[10];
    const float* b2  = (const float*)d_in[11];
    float* out = (float*)d_out;

    const int N = in_sizes[0] / DIM;
    const int E = in_sizes[1] / 2;
    const int O = in_sizes[11];      // 32
    const int* row = ei;
    const int* col = ei + E;

    // workspace carve-up (floats)
    float* f    = (float*)d_ws;
    float* dinv = f;
    float* T    = dinv + (size_t)((N + 63) & ~63);
    float* NB   = T   + (size_t)N * DIM;
    float* AGG  = NB  + (size_t)N * DIM;
    float* H    = AGG + (size_t)N * DIM;

    dim3 blk(256);
    dim3 ggrid((N + 127) / 128, DIM / 64);   // CT=4 -> 64 cols per block
    int scatter_blocks = (int)(((size_t)E * 64 + 255) / 256);
    int agg_blocks     = (int)(((size_t)N * 64 + 255) / 256);

    // degree -> dinv (in place)
    deg_init_kernel<<<(N + 255) / 256, blk, 0, stream>>>(dinv, N);
    deg_accum_kernel<<<(E + 255) / 256, blk, 0, stream>>>(row, dinv, E);
    dinv_kernel<<<(N + 255) / 256, blk, 0, stream>>>(dinv, N);

    // layer 0
    gemm_bias_wmma<4><<<ggrid, blk, 0, stream>>>(x, W0, b0, T, N, DIM);
    gemm_bias_wmma<4><<<ggrid, blk, 0, stream>>>(T, Wn0, bn0, NB, N, DIM);
    agg_init_kernel<<<agg_blocks, blk, 0, stream>>>(NB, dinv, AGG, N);
    edge_scatter_kernel<<<scatter_blocks, blk, 0, stream>>>(row, col, dinv, NB, AGG, E);
    combine_norm_relu_kernel<<<(N + 7) / 8, blk, 0, stream>>>(T, NB, AGG, H, N);

    // layer 1
    gemm_bias_wmma<4><<<ggrid, blk, 0, stream>>>(H, W1, b1, T, N, DIM);
    gemm_bias_wmma<4><<<ggrid, blk, 0, stream>>>(T, Wn1, bn1, NB, N, DIM);
    agg_init_kernel<<<agg_blocks, blk, 0, stream>>>(NB, dinv, AGG, N);
    edge_scatter_kernel<<<scatter_blocks, blk, 0, stream>>>(row, col, dinv, NB, AGG, E);
    combine_norm_relu_kernel<<<(N + 7) / 8, blk, 0, stream>>>(T, NB, AGG, H, N);

    // final 256 -> 32  (CT=2 -> 32 cols per block, grid.y = 1)
    dim3 fgrid((N + 127) / 128, O / 32);
    gemm_bias_wmma<2><<<fgrid, blk, 0, stream>>>(H, W2, b2, out, N, O);
}